// EPKuramotoClassifier_23270132810226
// MI455X (gfx1250) — compile-verified
//
#include <hip/hip_runtime.h>
#include <math.h>

// Problem dims (fixed by reference)
#define N_OSC   4096
#define BATCH   4
#define DIN     2048
#define NCLS    1000
#define NSTEPS  100
#define DT_C    0.01f
#define TWO_PI_C 6.283185307179586f

typedef __attribute__((ext_vector_type(2))) float v2f;
typedef __attribute__((ext_vector_type(8))) float v8f;

// ---------------------------------------------------------------------------
// 1) W = sigmoid(adjacency_logits) * (1 - eye) * coupling_matrix
// ---------------------------------------------------------------------------
__global__ void build_W(const float* __restrict__ logits,
                        const float* __restrict__ cm,
                        float* __restrict__ W) {
    size_t idx = (size_t)blockIdx.x * 256 + threadIdx.x;   // grid sized exactly
    int i = (int)(idx >> 12);
    int j = (int)(idx & (N_OSC - 1));
    float l  = logits[idx];
    float sg = 1.0f / (1.0f + expf(-l));
    float w  = sg * cm[idx];
    W[idx] = (i == j) ? 0.0f : w;
}

// ---------------------------------------------------------------------------
// 1b) xpad[16][DIN]: rows 0..3 = x, rows 4..15 = 0  (branch-free WMMA A loads)
// ---------------------------------------------------------------------------
__global__ void pad_x(const float* __restrict__ x, float* __restrict__ xpad) {
    int idx = blockIdx.x * 256 + threadIdx.x;              // 16*DIN total
    int row = idx / DIN;
    xpad[idx] = (row < BATCH) ? x[idx] : 0.0f;             // rows contiguous
}

// zero featspad rows 4..15 (12*2N floats); rows 0..3 written by snapshot
__global__ void zero_feats_tail(float* __restrict__ featspad) {
    int idx = blockIdx.x * 256 + threadIdx.x;
    featspad[BATCH * 2 * N_OSC + idx] = 0.0f;
}

// ---------------------------------------------------------------------------
// 2) ext = tanh(x @ enc_w + enc_b)   [4,2048]x[2048,4096] via fp32 WMMA 16x16x4
//    A fragment (16x4 f32, ISA layout): lanes 0-15 -> K=0,1 ; lanes 16-31 -> K=2,3
//    B fragment (4x16 f32): symmetric layout assumed (row striped across lanes)
//    One wave per 16-wide N tile; M padded 4 -> 16 with zeros (xpad).
// ---------------------------------------------------------------------------
__global__ void encoder_wmma(const float* __restrict__ xpad,
                             const float* __restrict__ enc_w,
                             const float* __restrict__ enc_b,
                             float* __restrict__ ext) {
    int wave = blockIdx.x * (blockDim.x >> 5) + (threadIdx.x >> 5);
    int lane = threadIdx.x & 31;
    int half = lane >> 4;          // 0: K=0,1   1: K=2,3
    int lr   = lane & 15;
    int n0   = wave * 16;

    const float* arow = xpad + (size_t)lr * DIN;
    v8f acc = {};
    for (int k = 0; k < DIN; k += 4) {
        int kk = k + 2 * half;
        v2f a, b;
        a[0] = arow[kk];
        a[1] = arow[kk + 1];
        b[0] = enc_w[(size_t)kk       * N_OSC + n0 + lr];
        b[1] = enc_w[(size_t)(kk + 1) * N_OSC + n0 + lr];
        acc = __builtin_amdgcn_wmma_f32_16x16x4_f32(
                  false, a, false, b, (short)0, acc, false, false);
    }
    // C/D layout: VGPR r, lanes 0-15 => (M=r, N=lane). Need rows 0..3 only.
    if (lane < 16) {
        float bias = enc_b[n0 + lane];
#pragma unroll
        for (int r = 0; r < BATCH; ++r)
            ext[r * N_OSC + n0 + lane] = tanhf(acc[r] + bias);
    }
}

// ---------------------------------------------------------------------------
// 3) init: phi = init_phases; s = sin(phi); c = cos(phi)
// ---------------------------------------------------------------------------
__global__ void init_phase(const float* __restrict__ init_phases,
                           float* __restrict__ phi,
                           float* __restrict__ s, float* __restrict__ c) {
    int j = blockIdx.x * 256 + threadIdx.x;
    float p = init_phases[j];
    phi[j] = p;
    s[j] = sinf(p);
    c[j] = cosf(p);
}

// ---------------------------------------------------------------------------
// 4) One Kuramoto Euler step. Wave-per-row fused dual GEMV (W stays hot in
//    the 192MB L2; 64MB per pass). s/c vectors staged in 32KB LDS per block.
//    coup_i = c_i*(W@s)_i - s_i*(W@c)_i ; phi_i = mod(phi_i + DT*(...), 2pi)
// ---------------------------------------------------------------------------
#define STEP_WAVES 16
__global__ __launch_bounds__(32 * STEP_WAVES)
void kuramoto_step(const float* __restrict__ W,
                   const float* __restrict__ omega,
                   const float* __restrict__ ext,      // this sample's drive
                   const float* __restrict__ Kp,       // coupling_strength[0]
                   float* __restrict__ phi,            // in-place update
                   const float* __restrict__ s_in,
                   const float* __restrict__ c_in,
                   float* __restrict__ s_out,
                   float* __restrict__ c_out) {
    __shared__ __align__(16) float s_sh[N_OSC];
    __shared__ __align__(16) float c_sh[N_OSC];
    for (int j = threadIdx.x; j < N_OSC; j += 32 * STEP_WAVES) {
        s_sh[j] = s_in[j];
        c_sh[j] = c_in[j];
    }
    __syncthreads();

    int wave = threadIdx.x >> 5;
    int lane = threadIdx.x & 31;
    int i = blockIdx.x * STEP_WAVES + wave;             // output row

    const float4* Wrow = (const float4*)(W + (size_t)i * N_OSC);
    const float4* s4   = (const float4*)s_sh;
    const float4* c4   = (const float4*)c_sh;

    float ws = 0.0f, wc = 0.0f;
#pragma unroll 4
    for (int q = lane; q < N_OSC / 4; q += 32) {
        float4 w  = Wrow[q];
        float4 sv = s4[q];
        float4 cv = c4[q];
        ws = fmaf(w.x, sv.x, ws); ws = fmaf(w.y, sv.y, ws);
        ws = fmaf(w.z, sv.z, ws); ws = fmaf(w.w, sv.w, ws);
        wc = fmaf(w.x, cv.x, wc); wc = fmaf(w.y, cv.y, wc);
        wc = fmaf(w.z, cv.z, wc); wc = fmaf(w.w, cv.w, wc);
    }
    // wave32 butterfly reduction
    for (int off = 16; off > 0; off >>= 1) {
        ws += __shfl_xor(ws, off, 32);
        wc += __shfl_xor(wc, off, 32);
    }
    if (lane == 0) {
        float K     = Kp[0];
        float scale = K * (1.0f / (float)N_OSC);
        float si = s_sh[i], ci = c_sh[i];
        float coup = ci * ws - si * wc;
        float p = phi[i] + DT_C * (omega[i] + scale * coup + ext[i]);
        p = fmodf(p, TWO_PI_C);
        if (p < 0.0f) p += TWO_PI_C;       // match jnp.mod semantics
        phi[i]   = p;
        s_out[i] = sinf(p);
        c_out[i] = cosf(p);
    }
}

// ---------------------------------------------------------------------------
// 5) featspad row b = concat(sin(phi), cos(phi))
// ---------------------------------------------------------------------------
__global__ void snapshot(const float* __restrict__ s,
                         const float* __restrict__ c,
                         float* __restrict__ feats_row) {
    int j = blockIdx.x * 256 + threadIdx.x;
    feats_row[j]         = s[j];
    feats_row[N_OSC + j] = c[j];
}

// ---------------------------------------------------------------------------
// 6) logits = feats @ readout_w + readout_b  [4,8192]x[8192,1000] via WMMA.
//    A comes from featspad (16 rows, 4..15 zero) -> unconditional loads.
//    B column index is clamped instead of predicated: out-of-range lanes
//    accumulate garbage that is never stored.
// ---------------------------------------------------------------------------
__global__ void readout_wmma(const float* __restrict__ featspad,
                             const float* __restrict__ w,
                             const float* __restrict__ bvec,
                             float* __restrict__ out) {
    int wave = blockIdx.x * (blockDim.x >> 5) + (threadIdx.x >> 5);
    const int ntiles = (NCLS + 15) / 16;   // 63
    if (wave >= ntiles) return;
    int lane = threadIdx.x & 31;
    int half = lane >> 4;
    int lr   = lane & 15;
    int n0   = wave * 16;
    int col  = n0 + lr;
    int colc = (col < NCLS) ? col : (NCLS - 1);   // clamped, branch-free loads

    const float* arow = featspad + (size_t)lr * (2 * N_OSC);
    v8f acc = {};
    for (int k = 0; k < 2 * N_OSC; k += 4) {
        int kk = k + 2 * half;
        v2f a, b;
        a[0] = arow[kk];
        a[1] = arow[kk + 1];
        b[0] = w[(size_t)kk       * NCLS + colc];
        b[1] = w[(size_t)(kk + 1) * NCLS + colc];
        acc = __builtin_amdgcn_wmma_f32_16x16x4_f32(
                  false, a, false, b, (short)0, acc, false, false);
    }
    if (lane < 16 && col < NCLS) {
        float bias = bvec[col];
#pragma unroll
        for (int r = 0; r < BATCH; ++r)
            out[r * NCLS + col] = acc[r] + bias;
    }
}

// ---------------------------------------------------------------------------
// Launcher
// ---------------------------------------------------------------------------
extern "C" void kernel_launch(void* const* d_in, const int* in_sizes, int n_in,
                              void* d_out, int out_size, void* d_ws, size_t ws_size,
                              hipStream_t stream) {
    (void)in_sizes; (void)n_in; (void)out_size; (void)ws_size;

    const float* x           = (const float*)d_in[0];
    const float* enc_w       = (const float*)d_in[1];
    const float* enc_b       = (const float*)d_in[2];
    const float* readout_w   = (const float*)d_in[3];
    const float* readout_b   = (const float*)d_in[4];
    const float* omega       = (const float*)d_in[5];
    const float* adj_logits  = (const float*)d_in[6];
    const float* coupling    = (const float*)d_in[7];
    const float* Kp          = (const float*)d_in[8];
    const float* init_phases = (const float*)d_in[9];
    float* out = (float*)d_out;

    // workspace layout (floats); W first so it warms L2 and stays resident
    float* ws_f     = (float*)d_ws;
    float* W        = ws_f;                                 // 4096*4096
    float* ext      = W    + (size_t)N_OSC * N_OSC;         // 4*4096
    float* phi      = ext  + (size_t)BATCH * N_OSC;         // 4096
    float* sA       = phi  + N_OSC;
    float* cA       = sA   + N_OSC;
    float* sB       = cA   + N_OSC;
    float* cB       = sB   + N_OSC;
    float* xpad     = cB   + N_OSC;                         // 16*2048
    float* featspad = xpad + 16 * DIN;                      // 16*8192

    build_W<<<(N_OSC * N_OSC) / 256, 256, 0, stream>>>(adj_logits, coupling, W);
    pad_x<<<(16 * DIN) / 256, 256, 0, stream>>>(x, xpad);
    zero_feats_tail<<<(12 * 2 * N_OSC) / 256, 256, 0, stream>>>(featspad);
    encoder_wmma<<<(N_OSC / 16) / 8, 256, 0, stream>>>(xpad, enc_w, enc_b, ext);
    init_phase<<<N_OSC / 256, 256, 0, stream>>>(init_phases, phi, sA, cA);

    float* sbuf[2] = { sA, sB };
    float* cbuf[2] = { cA, cB };
    int cur = 0;
    for (int b = 0; b < BATCH; ++b) {
        for (int t = 0; t < NSTEPS; ++t) {
            int nxt = cur ^ 1;
            kuramoto_step<<<N_OSC / STEP_WAVES, 32 * STEP_WAVES, 0, stream>>>(
                W, omega, ext + (size_t)b * N_OSC, Kp, phi,
                sbuf[cur], cbuf[cur], sbuf[nxt], cbuf[nxt]);
            cur = nxt;
        }
        snapshot<<<N_OSC / 256, 256, 0, stream>>>(
            sbuf[cur], cbuf[cur], featspad + (size_t)b * 2 * N_OSC);
    }
    readout_wmma<<<8, 256, 0, stream>>>(featspad, readout_w, readout_b, out);
}